// Auto_ODE_FHN_77532749627472
// MI455X (gfx1250) — compile-verified
//
#include <hip/hip_runtime.h>
#include <hip/hip_bf16.h>
#include <stdint.h>

// FitzHugh-Nagumo RK4, B trajectories x num_steps steps.
// Latency-bound serial recurrence in fp32; output (67 MB) drained via the
// CDNA5 Tensor Data Mover: LDS double-buffer staging + TENSOR_STORE_FROM_LDS,
// throttled with s_wait_tensorcnt.

typedef __attribute__((ext_vector_type(4))) unsigned int v4u;
typedef __attribute__((ext_vector_type(8))) int          v8i;
typedef __attribute__((ext_vector_type(4))) int          v4i;

#define TPB   128   // threads per block (4 wave32s)
#define CHUNK 32    // steps staged per TDM store

__global__ __launch_bounds__(TPB) void fhn_rk4_tdm_kernel(
    const float* __restrict__ x0, const float* __restrict__ y0,
    const float* __restrict__ pa, const float* __restrict__ pb,
    const float* __restrict__ pc, const int* __restrict__ pn,
    float* __restrict__ out, int B)
{
    // double-buffered staging: [buf][step-in-chunk][thread] = (x, y)
    // 2 * 32 * 128 * 8B = 64 KB LDS (WGP has 320 KB)
    __shared__ float2 stage[2][CHUNK][TPB];

    const int tid = threadIdx.x;
    const int gid = blockIdx.x * TPB + tid;

    const float a = *pa, b = *pb, c = *pc;
    const int nsteps = *pn;

    const float h    = 0.1f;
    const float half = 0.05f;
    const float h6   = h * (1.0f / 6.0f);
    const float c3   = c * (1.0f / 3.0f);   // f_x = c*(x+y) - c3*x^3
    const float nic  = -1.0f / c;           // f_y = nic*(x + b*y - a)

    float x = x0[gid];
    float y = y0[gid];

    auto fx = [&](float X, float Y) -> float {
        float t = X * X;
        return fmaf(c, X + Y, -(c3 * t) * X);
    };
    auto fy = [&](float X, float Y) -> float {
        return nic * fmaf(b, Y, X - a);
    };

    const bool tdm_ok = (B % TPB) == 0;

    int s = 0, buf = 0;
    while (s < nsteps) {
        const int cnt = (nsteps - s < CHUNK) ? (nsteps - s) : CHUNK;

        for (int j = 0; j < cnt; ++j) {
            // output row (s+j) is the state BEFORE this integration step
            stage[buf][j][tid] = make_float2(x, y);

            float k1x = fx(x, y),   k1y = fy(x, y);
            float xa  = fmaf(k1x, half, x), ya = fmaf(k1y, half, y);
            float k2x = fx(xa, ya), k2y = fy(xa, ya);
            float xb  = fmaf(k2x, half, x), yb = fmaf(k2y, half, y);
            float k3x = fx(xb, yb), k3y = fy(xb, yb);
            float xc  = fmaf(k3x, h, x),    yc = fmaf(k3y, h, y);
            float k4x = fx(xc, yc), k4y = fy(xc, yc);
            x = fmaf(h6, k1x + k4x + 2.0f * (k2x + k3x), x);
            y = fmaf(h6, k1y + k4y + 2.0f * (k2y + k3y), y);
        }

        __syncthreads();   // all lanes' ds_store_b64 visible in LDS

        if (tdm_ok && cnt == CHUNK) {
            if (tid < 32) {  // wave 0 drives the TDM engine for the block
                // ---- Tensor DMA descriptor (2D tile, data_size = 4B) ----
                // tile: 256 dwords (128 lanes * float2) x 32 rows
                // memory row stride: B*2 dwords (one full output row)
                unsigned lds_off = (unsigned)(uintptr_t)(&stage[buf][0][0]);
                unsigned long long gaddr =
                    (unsigned long long)(uintptr_t)out +
                    ((unsigned long long)s * (unsigned)B +
                     (unsigned long long)blockIdx.x * TPB) * 8ull;

                v4u g0 = {
                    1u,                                   // count=1 (valid), no gather
                    lds_off,                              // lds_addr
                    (unsigned)gaddr,                      // global_addr[31:0]
                    ((unsigned)(gaddr >> 32) & 0x01FFFFFFu) | 0x80000000u // addr[56:32] | type=2
                };
                const int row_dw = TPB * 2;               // 256 dwords per tile row
                v8i g1 = {
                    (int)(2u << 16),                      // data_size=4B; wg_mask=0
                    (int)((unsigned)row_dw << 16),        // tensor_dim0[15:0]
                    (int)((unsigned)CHUNK << 16),         // tensor_dim0[31:16]=0 | tensor_dim1[15:0]
                    (int)((unsigned)row_dw << 16),        // tensor_dim1[31:16]=0 | tile_dim0
                    CHUNK,                                // tile_dim1 | tile_dim2=0
                    B * 2,                                // tensor_dim0_stride[31:0]
                    0,                                    // stride0[47:32] | stride1[15:0]
                    0
                };
                v4i g2 = {0, 0, 0, 0};                    // 2D tile: dims 2/3 unused
                v4i g3 = {0, 0, 0, 0};
                v8i g4 = {0, 0, 0, 0, 0, 0, 0, 0};        // unused trailing group (6-arg builtin)
                __builtin_amdgcn_tensor_store_from_lds(g0, g1, g2, g3, g4, 0);
                // double-buffer throttle: oldest TDM (2 chunks back, the buffer
                // we refill next) must be complete before the block reuses it
                __builtin_amdgcn_s_wait_tensorcnt(1);
            }
        } else {
            // tail (num_steps % CHUNK != 0): plain coalesced float2 stores
            float2* o = (float2*)out;
            for (int j = 0; j < cnt; ++j)
                o[(size_t)(s + j) * (unsigned)B + gid] = stage[buf][j][tid];
        }

        __syncthreads();   // release all waves for the next chunk
        s += cnt;
        buf ^= 1;
    }
    // s_endpgm performs an implicit wait-idle, draining any outstanding TDM op.
}

extern "C" void kernel_launch(void* const* d_in, const int* in_sizes, int n_in,
                              void* d_out, int out_size, void* d_ws, size_t ws_size,
                              hipStream_t stream) {
    const float* x0 = (const float*)d_in[0];
    const float* y0 = (const float*)d_in[1];
    const float* pa = (const float*)d_in[2];
    const float* pb = (const float*)d_in[3];
    const float* pc = (const float*)d_in[4];
    const int*   pn = (const int*)d_in[5];
    float* out = (float*)d_out;

    const int B = in_sizes[0];           // 4096 trajectories
    dim3 block(TPB);
    dim3 grid((B + TPB - 1) / TPB);      // 32 blocks = 128 wave32 waves

    fhn_rk4_tdm_kernel<<<grid, block, 0, stream>>>(x0, y0, pa, pb, pc, pn, out, B);
}